// AI4DEM_34531537059900
// MI455X (gfx1250) — compile-verified
//
#include <hip/hip_runtime.h>
#include <cmath>

#define AS1 __attribute__((address_space(1)))
#define AS3 __attribute__((address_space(3)))

namespace {

constexpr int Zd = 256, Yd = 128, Xd = 128;
constexpr int N  = Zd * Yd * Xd;

constexpr float CELL  = 0.003f;
constexpr float Dp    = CELL;           // d == cell_size
constexpr float TWO_D = 2.0f * Dp;
constexpr float KNc   = 10000.0f;
constexpr float EPSc  = 0.0001f;
constexpr float DTc   = 0.0001f;
constexpr float LXf   = Xd * CELL;      // 0.384
constexpr float LYf   = Yd * CELL;      // 0.384
constexpr float LZf   = Zd * CELL;      // 0.768

// Tile: 32 (x, coalesced) x 4 (y) x 4 (z) = 512 threads = 16 wave32s.
constexpr int TX = 32, TY = 4, TZ = 4;
constexpr int HX = TX + 4, HY = TY + 4, HZ = TZ + 4;   // 36 x 8 x 8 halo tile
constexpr int NTILE = HX * HY * HZ;                    // 2304 elements/field
constexpr int NTHREADS = TX * TY * TZ;                 // 512

// ---- CDNA5 async global->LDS copy (per-lane 4B gather into LDS) ----------
__device__ __forceinline__ void async_copy_b32(const float* g, float* l) {
#if __has_builtin(__builtin_amdgcn_global_load_async_to_lds_b32)
  // (AS1 int* src, AS3 int* dst, imm off, imm cpol); LDS byte offset ==
  // low 32 bits of the generic pointer (ISA 10.2 aperture rule).
  __builtin_amdgcn_global_load_async_to_lds_b32(
      (AS1 int*)(unsigned long long)g,
      (AS3 int*)(unsigned)(unsigned long long)l,
      0, 0);
#else
  unsigned loff = (unsigned)(unsigned long long)l;       // addr[31:0] = LDS offset
  unsigned long long gaddr = (unsigned long long)g;
  asm volatile("global_load_async_to_lds_b32 %0, %1, off"
               :: "v"(loff), "v"(gaddr) : "memory");
#endif
}

__device__ __forceinline__ void wait_async0() {
#if __has_builtin(__builtin_amdgcn_s_wait_asynccnt)
  __builtin_amdgcn_s_wait_asynccnt(0);
#else
  asm volatile("s_wait_asynccnt 0" ::: "memory");
#endif
}

__global__ __launch_bounds__(NTHREADS)
void dem_step_kernel(const float* __restrict__ xg,  const float* __restrict__ yg,
                     const float* __restrict__ zg,  const float* __restrict__ vxg,
                     const float* __restrict__ vyg, const float* __restrict__ vzg,
                     const float* __restrict__ mg,  float* __restrict__ out,
                     float ETA, float ETA_WALL, float C /* = DT/MASS */)
{
  // AoS tiles: one ds_load_b128 per vector quantity per neighbor.
  __shared__ float4 smp[NTILE];   // {x, y, z, -}   36864 B
  __shared__ float4 smv[NTILE];   // {vx, vy, vz, -} 36864 B

  const int x0 = blockIdx.x * TX;
  const int y0 = blockIdx.y * TY;
  const int z0 = blockIdx.z * TZ;
  const int tid = (threadIdx.z * TY + threadIdx.y) * TX + threadIdx.x;

  // ---- stage tile+halo into LDS with async DMA, periodic wrap ----
  for (int t = tid; t < NTILE; t += NTHREADS) {
    const int lx = t % HX;
    const int r  = t / HX;
    const int ly = r % HY;
    const int lz = r / HY;
    const int gx = (x0 + lx - 2 + Xd) & (Xd - 1);
    const int gy = (y0 + ly - 2 + Yd) & (Yd - 1);
    const int gz = (z0 + lz - 2 + Zd) & (Zd - 1);
    const int gi = (gz * Yd + gy) * Xd + gx;
    float* p = (float*)&smp[t];
    float* v = (float*)&smv[t];
    async_copy_b32(xg  + gi, p + 0);
    async_copy_b32(yg  + gi, p + 1);
    async_copy_b32(zg  + gi, p + 2);
    async_copy_b32(vxg + gi, v + 0);
    async_copy_b32(vyg + gi, v + 1);
    async_copy_b32(vzg + gi, v + 2);
  }
  wait_async0();
  __syncthreads();

  // ---- center values ----
  const int cx = threadIdx.x + 2, cy = threadIdx.y + 2, cz = threadIdx.z + 2;
  const int cidx = (cz * HY + cy) * HX + cx;
  const float4 pc = smp[cidx];
  const float4 vc = smv[cidx];
  const float xc = pc.x, yc = pc.y, zc = pc.z;
  const float vxc = vc.x, vyc = vc.y, vzc = vc.z;

  // ---- 125-neighbor contact + damping forces ----
  // sum over roll shifts s in [-2,2]^3 of (g - roll(g,s)) == sum over
  // neighbor offsets o in [-2,2]^3 of (g[c] - g[c+o]) with periodic wrap.
  float fpx = 0.0f, fpy = 0.0f, fpz = 0.0f;
  for (int oz = -2; oz <= 2; ++oz) {
    for (int oy = -2; oy <= 2; ++oy) {
      const int base = ((cz + oz) * HY + (cy + oy)) * HX + cx;
#pragma unroll
      for (int ox = -2; ox <= 2; ++ox) {
        const int n = base + ox;
        const float4 pn = smp[n];          // ds_load_b128
        const float4 vn = smv[n];          // ds_load_b128
        const float dx = xc - pn.x;
        const float dy = yc - pn.y;
        const float dz = zc - pn.z;
        const float d2   = dx * dx + dy * dy + dz * dz;
        const float dist = __builtin_amdgcn_sqrtf(d2);
        const float den  = fmaxf(EPSc, dist);
        const float inv  = __builtin_amdgcn_rcpf(den);
        const float dvx = vxc - vn.x;
        const float dvy = vyc - vn.y;
        const float dvz = vzc - vn.z;
        const float dot = dvx * dx + dvy * dy + dvz * dz;
        // f = KN*(dist-2d)/den + ETA*(dot/den)/den, factored by inv once
        float f = inv * (KNc * (dist - TWO_D) + ETA * (dot * inv));
        f = (dist < TWO_D) ? f : 0.0f;     // hit mask (f*_c + f*_d combined)
        fpx += f * dx;
        fpy += f * dy;
        fpz += f * dz;
      }
    }
  }

  // ---- wall forces + integration ----
  const int gx = x0 + threadIdx.x, gy = y0 + threadIdx.y, gz = z0 + threadIdx.z;
  const int idx = (gz * Yd + gy) * Xd + gx;
  const float m = mg[idx];

  const float il  = (xc != 0.0f && xc < Dp)        ? 1.0f : 0.0f;
  const float ir  = (xc > LXf - TWO_D)             ? 1.0f : 0.0f;
  const float ib  = (yc != 0.0f && yc < Dp)        ? 1.0f : 0.0f;
  const float it  = (yc > LYf - TWO_D)             ? 1.0f : 0.0f;
  const float ifw = (zc != 0.0f && zc < 3.0f * Dp) ? 1.0f : 0.0f;
  const float ibk = (zc > LZf - TWO_D)             ? 1.0f : 0.0f;

  // vx_n = vx + C*(-fx_br + fx_bl - fx_c - fx_d - fx_ld - fx_rd)*mask  (mask^2 == mask)
  const float ax = KNc * il  * (Dp - xc)        - KNc * ir  * (xc - LXf + TWO_D)
                 - fpx - ETA_WALL * vxc * (il + ir);
  const float ay = KNc * ib  * (Dp - yc)        - KNc * it  * (yc - LYf + TWO_D)
                 - fpy - ETA_WALL * vyc * (ib + it);
  const float az = KNc * ifw * (3.0f * Dp - zc) - KNc * ibk * (zc - LZf + TWO_D)
                 - fpz - ETA_WALL * vzc * (ifw + ibk);

  const float vxn = vxc + C * ax * m;
  const float vyn = vyc + C * ay * m;
  const float vzn = vzc + DTc * (-9.8f) * m + C * az * m;
  const float xn  = xc + DTc * vxn;
  const float yn  = yc + DTc * vyn;
  const float zn  = zc + DTc * vzn;

  out[0 * N + idx] = xn;
  out[1 * N + idx] = yn;
  out[2 * N + idx] = zn;
  out[3 * N + idx] = vxn;
  out[4 * N + idx] = vyn;
  out[5 * N + idx] = vzn;
}

} // anonymous namespace

extern "C" void kernel_launch(void* const* d_in, const int* in_sizes, int n_in,
                              void* d_out, int out_size, void* d_ws, size_t ws_size,
                              hipStream_t stream) {
  (void)in_sizes; (void)n_in; (void)out_size; (void)d_ws; (void)ws_size;

  const float* xg  = (const float*)d_in[0];
  const float* yg  = (const float*)d_in[1];
  const float* zg  = (const float*)d_in[2];
  const float* vxg = (const float*)d_in[3];
  const float* vyg = (const float*)d_in[4];
  const float* vzg = (const float*)d_in[5];
  const float* mg  = (const float*)d_in[6];
  float* out = (float*)d_out;

  // Constants exactly as the reference computes them (double precision).
  const double alpha = 0.6931471805599453 / 3.141592653589793;   // -ln(0.5)/pi
  const double gam   = alpha / std::sqrt(alpha * alpha + 1.0);
  const double mass  = 4.0 / 3.0 * 3.1415 * (0.003 * 0.003 * 0.003) * 1592.0;
  const float eta      = (float)(2.0 * gam * std::sqrt(10000.0 * mass / 2.0));
  const float eta_wall = (float)(2.0 * gam * std::sqrt(10000.0 * mass));
  const float c        = (float)(0.0001 / mass);                  // DT / MASS

  dim3 grid(Xd / TX, Yd / TY, Zd / TZ);   // 4 x 32 x 64
  dim3 block(TX, TY, TZ);                 // 32 x 4 x 4 = 512 threads (16 waves)
  hipLaunchKernelGGL(dem_step_kernel, grid, block, 0, stream,
                     xg, yg, zg, vxg, vyg, vzg, mg, out, eta, eta_wall, c);
}